// CKGGCN_19859928777171
// MI455X (gfx1250) — compile-verified
//
#include <hip/hip_runtime.h>
#include <math.h>

// CKGGCN aggregation for MI455X (gfx1250, wave32).
// Key optimization: project entity_emb @ W_Q once per entity (not per edge)
// with V_WMMA_F32_16X16X4_F32; all segment ops via L2-resident f32 atomics.

typedef __attribute__((ext_vector_type(2))) float v2f;
typedef __attribute__((ext_vector_type(8))) float v8f;

#define D 128          // embedding dim
#define LDSPITCH 132   // 128 + 4 pad: lane stride hits distinct banks

__device__ __forceinline__ unsigned order_key(float x) {
    unsigned b = __float_as_uint(x);
    return (b & 0x80000000u) ? ~b : (b | 0x80000000u);
}
__device__ __forceinline__ float order_val(unsigned k) {
    unsigned b = (k & 0x80000000u) ? (k & 0x7FFFFFFFu) : ~k;
    return __uint_as_float(b);
}

// ---------------- init: d_out = emb/3, entA = entity_emb copy ----------------
__global__ void init_user_k(const float* __restrict__ user_emb,
                            float* __restrict__ out_user, int n) {
    int i = blockIdx.x * blockDim.x + threadIdx.x;
    if (i < n) out_user[i] = user_emb[i] * (1.0f / 3.0f);
}
__global__ void init_entity_k(const float* __restrict__ ent_emb,
                              float* __restrict__ out_ent,
                              float* __restrict__ entA, int n) {
    int i = blockIdx.x * blockDim.x + threadIdx.x;
    if (i < n) {
        float v = ent_emb[i];
        out_ent[i] = v * (1.0f / 3.0f);
        entA[i] = v;
    }
}

// ---------------- P = ent_in @ W_Q via V_WMMA_F32_16X16X4_F32 ----------------
// One block = one 16-row M tile; 8 waves cover the 8 N tiles (128 cols).
// A tile (16x128 f32) staged in LDS; W_Q (64 KB) stays hot in L2.
__global__ __launch_bounds__(256)
void proj_wmma_k(const float* __restrict__ ent, const float* __restrict__ WQ,
                 float* __restrict__ P, int NE) {
    __shared__ float lds[16 * LDSPITCH];
    const int row0 = blockIdx.x * 16;
    const int t = threadIdx.x;

    // cooperative load: 256 threads x 8 contiguous floats = 16x128 tile
    #pragma unroll
    for (int j = 0; j < 8; ++j) {
        int idx = t * 8 + j;
        int r = idx >> 7, c = idx & 127;
        float v = (row0 + r < NE) ? ent[(size_t)(row0 + r) * D + c] : 0.0f;
        lds[r * LDSPITCH + c] = v;
    }
    __syncthreads();

    const int wave = t >> 5, lane = t & 31;
    const int ntile = wave;                 // N tile 0..7
    const int mi    = lane & 15;            // M index for A / N index for B
    const int koff  = (lane >> 4) << 1;     // lanes 0-15: K0/K1, lanes 16-31: K2/K3
    const float* bcol = WQ + ntile * 16 + mi;   // W_Q row-major [k][n]

    v8f acc = {};
    #pragma unroll
    for (int k0 = 0; k0 < D; k0 += 4) {
        v2f a, b;
        a.x = lds[mi * LDSPITCH + k0 + koff];
        a.y = lds[mi * LDSPITCH + k0 + koff + 1];
        b.x = bcol[(size_t)(k0 + koff) * D];
        b.y = bcol[(size_t)(k0 + koff + 1) * D];
        acc = __builtin_amdgcn_wmma_f32_16x16x4_f32(
            /*neg_a=*/false, a, /*neg_b=*/false, b,
            /*c_mod=*/(short)0, acc, /*reuse_a=*/false, /*reuse_b=*/false);
    }

    // C/D layout: VGPR r -> M=r (lanes 0-15) or M=r+8 (lanes 16-31), N=lane%16
    const int rbase = row0 + ((lane >> 4) << 3);
    const int col = ntile * 16 + mi;
    #pragma unroll
    for (int r = 0; r < 8; ++r) {
        int rr = rbase + r;
        if (rr < NE) P[(size_t)rr * D + col] = acc[r];
    }
}

// ---------------- per-edge score + atomic segment-max ----------------
// One wave per edge; lanes 0-15 reduce head 0 (dims 0..63), lanes 16-31 head 1.
__global__ __launch_bounds__(256)
void edge_score_k(const float* __restrict__ P, const int* __restrict__ eidx,
                  const int* __restrict__ etype, const float* __restrict__ rel,
                  float* __restrict__ escore, unsigned* __restrict__ maxkey, int E) {
    int gw = (int)((blockIdx.x * blockDim.x + threadIdx.x) >> 5);
    if (gw >= E) return;
    int lane = threadIdx.x & 31;
    int h = eidx[gw], tl = eidx[E + gw];
    int r = etype[gw] - 1;
    const float* ph = P + (size_t)h * D;
    const float* pt = P + (size_t)tl * D;
    const float* pr = rel + (size_t)r * D;
    int base = lane * 4;
    float s = 0.0f;
    #pragma unroll
    for (int j = 0; j < 4; ++j) s += ph[base + j] * pt[base + j] * pr[base + j];
    s += __shfl_xor(s, 8, 16);
    s += __shfl_xor(s, 4, 16);
    s += __shfl_xor(s, 2, 16);
    s += __shfl_xor(s, 1, 16);
    if ((lane & 15) == 0) {
        int hh = lane >> 4;
        s *= 0.125f;  // 1/sqrt(d_k), d_k = 64
        escore[(size_t)gw * 2 + hh] = s;
        atomicMax(&maxkey[(size_t)h * 2 + hh], order_key(s));
    }
}

// ---------------- exp(score - max) + atomic segment-sum ----------------
__global__ __launch_bounds__(256)
void edge_exp_k(const int* __restrict__ eidx, float* __restrict__ escore,
                const unsigned* __restrict__ maxkey, float* __restrict__ expsum,
                int E) {
    int i = blockIdx.x * blockDim.x + threadIdx.x;
    if (i >= 2 * E) return;
    int e = i >> 1, hh = i & 1;
    int h = eidx[e];
    float m = order_val(maxkey[(size_t)h * 2 + hh]);
    float v = __expf(escore[i] - m);
    escore[i] = v;
    unsafeAtomicAdd(&expsum[(size_t)h * 2 + hh], v);
}

// ---------------- attention-weighted scatter: ent_out[head] += attn*tail*rel -
__global__ __launch_bounds__(256)
void edge_agg_k(const float* __restrict__ ent_in, const int* __restrict__ eidx,
                const int* __restrict__ etype, const float* __restrict__ rel,
                const float* __restrict__ escore, const float* __restrict__ expsum,
                float* __restrict__ ent_out, int E) {
    int gw = (int)((blockIdx.x * blockDim.x + threadIdx.x) >> 5);
    if (gw >= E) return;
    int lane = threadIdx.x & 31;
    int h = eidx[gw], tl = eidx[E + gw];
    int r = etype[gw] - 1;
    int hh = lane >> 4;  // dims 0..63 -> head 0, 64..127 -> head 1
    float attn = escore[(size_t)gw * 2 + hh] / expsum[(size_t)h * 2 + hh];
    int base = lane * 4;
    const float* pt = ent_in + (size_t)tl * D + base;
    const float* pr = rel + (size_t)r * D + base;
    float* po = ent_out + (size_t)h * D + base;
    #pragma unroll
    for (int j = 0; j < 4; ++j)
        unsafeAtomicAdd(&po[j], pt[j] * pr[j] * attn);
}

// ---------------- row L2-normalize + accumulate /3 into d_out entity --------
__global__ __launch_bounds__(256)
void l2norm_acc_k(float* __restrict__ ent_out, float* __restrict__ dst, int NE) {
    int gw = (int)((blockIdx.x * blockDim.x + threadIdx.x) >> 5);
    if (gw >= NE) return;
    int lane = threadIdx.x & 31;
    int base = lane * 4;
    float* p = ent_out + (size_t)gw * D + base;
    float v0 = p[0], v1 = p[1], v2 = p[2], v3 = p[3];
    float ss = v0 * v0 + v1 * v1 + v2 * v2 + v3 * v3;
    ss += __shfl_xor(ss, 16, 32);
    ss += __shfl_xor(ss, 8, 32);
    ss += __shfl_xor(ss, 4, 32);
    ss += __shfl_xor(ss, 2, 32);
    ss += __shfl_xor(ss, 1, 32);
    float scale = 1.0f / fmaxf(sqrtf(ss), 1e-12f);
    v0 *= scale; v1 *= scale; v2 *= scale; v3 *= scale;
    p[0] = v0; p[1] = v1; p[2] = v2; p[3] = v3;
    float* q = dst + (size_t)gw * D + base;
    q[0] += v0 * (1.0f / 3.0f);
    q[1] += v1 * (1.0f / 3.0f);
    q[2] += v2 * (1.0f / 3.0f);
    q[3] += v3 * (1.0f / 3.0f);
}

// ---------------- user side: d_out_user[u] += (w/3) * ent_in[i] -------------
__global__ __launch_bounds__(256)
void user_agg_k(const float* __restrict__ ent_in, const int* __restrict__ iedge,
                const float* __restrict__ iw, float* __restrict__ out_user, int EI) {
    int gw = (int)((blockIdx.x * blockDim.x + threadIdx.x) >> 5);
    if (gw >= EI) return;
    int lane = threadIdx.x & 31;
    int u = iedge[gw], it = iedge[EI + gw];
    float w = iw[gw] * (1.0f / 3.0f);
    int base = lane * 4;
    const float* pi = ent_in + (size_t)it * D + base;
    float* pu = out_user + (size_t)u * D + base;
    #pragma unroll
    for (int j = 0; j < 4; ++j)
        unsafeAtomicAdd(&pu[j], w * pi[j]);
}

extern "C" void kernel_launch(void* const* d_in, const int* in_sizes, int n_in,
                              void* d_out, int out_size, void* d_ws, size_t ws_size,
                              hipStream_t stream) {
    const float* user_emb = (const float*)d_in[1];
    const float* ent_emb  = (const float*)d_in[2];
    const int*   iedge    = (const int*)d_in[3];
    const float* iw       = (const float*)d_in[4];
    const int*   eidx     = (const int*)d_in[5];
    const int*   etype    = (const int*)d_in[6];
    const float* rel      = (const float*)d_in[7];
    const float* WQ       = (const float*)d_in[8];

    const int NU = in_sizes[1] / D;   // 30000
    const int NE = in_sizes[2] / D;   // 60000
    const int EI = in_sizes[4];       // 800000
    const int E  = in_sizes[6];       // 600000

    float* out_user = (float*)d_out;
    float* out_ent  = out_user + (size_t)NU * D;

    // workspace carve-out (~98 MB total)
    char* ws = (char*)d_ws;
    size_t off = 0;
    auto carve = [&](size_t bytes) -> void* {
        void* p = ws + off;
        off = (off + bytes + 255) & ~(size_t)255;
        return p;
    };
    float*    P    = (float*)   carve((size_t)NE * D * sizeof(float));
    float*    entA = (float*)   carve((size_t)NE * D * sizeof(float));
    float*    entB = (float*)   carve((size_t)NE * D * sizeof(float));
    unsigned* maxk = (unsigned*)carve((size_t)NE * 2 * sizeof(unsigned));
    float*    esum = (float*)   carve((size_t)NE * 2 * sizeof(float));
    float*    esc  = (float*)   carve((size_t)E  * 2 * sizeof(float));

    init_user_k<<<(NU * D + 255) / 256, 256, 0, stream>>>(user_emb, out_user, NU * D);
    init_entity_k<<<(NE * D + 255) / 256, 256, 0, stream>>>(ent_emb, out_ent, entA, NE * D);

    float* ent_in = entA;
    float* ent_out = entB;
    for (int layer = 0; layer < 2; ++layer) {  // layers_num fixed at 2 in setup
        hipMemsetAsync(maxk, 0, (size_t)NE * 2 * sizeof(unsigned), stream);
        hipMemsetAsync(esum, 0, (size_t)NE * 2 * sizeof(float), stream);
        hipMemsetAsync(ent_out, 0, (size_t)NE * D * sizeof(float), stream);

        proj_wmma_k<<<(NE + 15) / 16, 256, 0, stream>>>(ent_in, WQ, P, NE);
        edge_score_k<<<(E + 7) / 8, 256, 0, stream>>>(P, eidx, etype, rel, esc, maxk, E);
        edge_exp_k<<<(2 * E + 255) / 256, 256, 0, stream>>>(eidx, esc, maxk, esum, E);
        edge_agg_k<<<(E + 7) / 8, 256, 0, stream>>>(ent_in, eidx, etype, rel, esc, esum, ent_out, E);
        l2norm_acc_k<<<(NE + 7) / 8, 256, 0, stream>>>(ent_out, out_ent, NE);
        user_agg_k<<<(EI + 7) / 8, 256, 0, stream>>>(ent_in, iedge, iw, out_user, EI);

        float* tmp = ent_in; ent_in = ent_out; ent_out = tmp;
    }
}